// Sobelxy_6493990552041
// MI455X (gfx1250) — compile-verified
//
#include <hip/hip_runtime.h>
#include <hip/hip_bf16.h>
#include <stdint.h>

// TDM descriptor SGPR-group vector types (clang-23 / therock-10.0 6-arg builtin form:
//   __builtin_amdgcn_tensor_load_to_lds(uint32x4 g0, int32x8 g1, int32x4 g2, int32x4 g3,
//                                       int32x8 g4, i32 cpol))
typedef unsigned int v4u __attribute__((ext_vector_type(4)));
typedef int          v8i __attribute__((ext_vector_type(8)));
typedef int          v4i __attribute__((ext_vector_type(4)));

#define TH     32            // output rows per workgroup
#define W      256           // image width (floats)
#define H      256           // image height
#define RS     258           // LDS row stride in floats: 256 data + 2 TDM-pad guard cols
#define SLACK  2             // leading guard (covers row0 col -1 access)
#define LROWS  (TH + 2)      // rows staged in LDS (tile + top/bottom halo)

__global__ __launch_bounds__(256) void sobel_tdm_kernel(const float* __restrict__ x,
                                                        float* __restrict__ out) {
    __shared__ float smem[SLACK + LROWS * RS];   // 8774 floats ~= 34.3 KB of 320 KB/WGP

    const int rb    = blockIdx.x;   // row-block 0..7
    const int plane = blockIdx.y;   // 0..1023 (B*C)
    const int tid   = threadIdx.x;  // 0..255
    const int r0    = rb * TH;

    // Clamp the halo at image top/bottom; TDM loads only the valid rows.
    const int top_clip    = (r0 == 0) ? 1 : 0;
    const int start_row   = r0 - 1 + top_clip;          // first global row loaded
    const int avail       = H - start_row;
    const int want        = LROWS - top_clip;
    const int nrows       = (want < avail) ? want : avail;
    const int first_local = top_clip;                   // local LDS row of first loaded row
    const int last_filled = first_local + nrows - 1;

    // --- Wave 0 issues the TDM 2D tile load FIRST so the DMA overlaps the LDS zero-fill ---
    if (tid < 32) {
        const uint64_t gaddr = (uint64_t)(uintptr_t)(x + (size_t)plane * (H * W)
                                                       + (size_t)start_row * W);
        const uint32_t laddr = (uint32_t)(uintptr_t)(&smem[SLACK + first_local * RS]);

        v4u g0;
        g0[0] = 1u;                                         // count=1 (valid), user mode
        g0[1] = laddr;                                      // lds_addr
        g0[2] = (uint32_t)(gaddr & 0xFFFFFFFFu);            // global_addr[31:0]
        g0[3] = (uint32_t)((gaddr >> 32) & 0x01FFFFFFu)     // global_addr[56:32]
              | (2u << 30);                                 // type=2 ("image")

        const uint32_t td0 = W;                // tensor_dim0 (elements)
        const uint32_t td1 = (uint32_t)nrows;  // tensor_dim1 = rows we may touch
        const uint32_t tl0 = W;                // tile_dim0
        const uint32_t tl1 = (uint32_t)nrows;  // tile_dim1
        const uint32_t str0 = W;               // tensor_dim0_stride (elements)

        v8i g1;
        g1[0] = (int)((2u << 16)        // data_size = 2 -> 4 bytes
                    | (1u << 20)        // pad_enable
                    | (7u << 22)        // pad_interval = 7 -> every 256 dwords (one row)
                    | (1u << 25));      // pad_amount = 1 -> 2 dwords (guard cols)
        g1[1] = (int)((td0 & 0xFFFFu) << 16);                         // tensor_dim0[15:0]
        g1[2] = (int)(((td0 >> 16) & 0xFFFFu) | ((td1 & 0xFFFFu) << 16));
        g1[3] = (int)(((td1 >> 16) & 0xFFFFu) | (tl0 << 16));         // tile_dim0
        g1[4] = (int)(tl1 & 0xFFFFu);                                 // tile_dim1, tile_dim2=0
        g1[5] = (int)str0;                                            // dim0_stride[31:0]
        g1[6] = 0;                                                    // stride0[47:32]|stride1[15:0]
        g1[7] = 0;                                                    // stride1[47:16]

        v4i g2 = {0, 0, 0, 0};             // 2D tile: dims 2/3 unused
        v4i g3 = {0, 0, 0, 0};
        v8i g4 = {0, 0, 0, 0, 0, 0, 0, 0}; // extra group required by 6-arg builtin (zero-filled)

        __builtin_amdgcn_tensor_load_to_lds(g0, g1, g2, g3, g4, 0);
    }

    // --- Pre-zero guard columns / slack / unloaded halo rows (disjoint from TDM writes) ---
    if (tid < SLACK) smem[tid] = 0.0f;
    if (tid < LROWS * 2) {
        const int row = tid >> 1;
        const int col = W + (tid & 1);                  // the 2 pad columns TDM skips
        smem[SLACK + row * RS + col] = 0.0f;
    }
    if (top_clip) smem[SLACK + 0 * RS + tid] = 0.0f;    // zero top halo row (image edge)
    for (int rz = last_filled + 1; rz < LROWS; ++rz)    // zero bottom halo row(s)
        smem[SLACK + rz * RS + tid] = 0.0f;

    // Wave 0 drains its TENSORcnt; waves 1-7 pass through instantly (count already 0).
    __builtin_amdgcn_s_wait_tensorcnt(0);
    __syncthreads();

    // --- Compute: one column per thread, sliding 3x3 register window over 32 rows ---
    const int c = tid;
    const float* s = &smem[SLACK + c];
    // s[-1] at c==0 lands on zeroed slack/guard; s[+1] at c==255 lands on zeroed pad col.
    float tl = s[-1],      tc = s[0],  tr = s[1];
    float ml = s[RS - 1],  mc = s[RS], mr = s[RS + 1];

    const size_t obase = (size_t)plane * (H * W) + (size_t)r0 * W + (size_t)c;

#pragma unroll 8
    for (int rr = 0; rr < TH; ++rr) {
        const int bi = (rr + 2) * RS;
        const float bl = s[bi - 1], bc = s[bi], br = s[bi + 1];

        const float gx = (tr - tl) + 2.0f * (mr - ml) + (br - bl);
        const float gy = (tl + 2.0f * tc + tr) - (bl + 2.0f * bc + br);

        out[obase + (size_t)rr * W] = __builtin_amdgcn_sqrtf(gx * gx + gy * gy + 1e-8f);

        tl = ml; tc = mc; tr = mr;    // slide window down one row
        ml = bl; mc = bc; mr = br;
    }
}

extern "C" void kernel_launch(void* const* d_in, const int* in_sizes, int n_in,
                              void* d_out, int out_size, void* d_ws, size_t ws_size,
                              hipStream_t stream) {
    (void)in_sizes; (void)n_in; (void)out_size; (void)d_ws; (void)ws_size;
    const float* x  = (const float*)d_in[0];
    float*       out = (float*)d_out;

    dim3 grid(H / TH, 16 * 64);   // 8 row-blocks x 1024 planes
    dim3 block(256);              // 8 wave32 per workgroup
    sobel_tdm_kernel<<<grid, block, 0, stream>>>(x, out);
}